// GaussPyramid_33715493273651
// MI455X (gfx1250) — compile-verified
//
#include <hip/hip_runtime.h>

typedef __attribute__((ext_vector_type(2))) float v2f;
typedef __attribute__((ext_vector_type(8))) float v8f;

#define KS    5
#define PAD   2
#define OTH   16              // output tile rows per block
#define OTW   32              // output tile cols per block (2 waves x 16)
#define TINH  (2*OTH + 3)     // 35 input/intermediate rows
#define TINW  (2*OTW + 3)     // 67 input cols
#define INSTR 68              // sIn row stride
#define HSTR  40              // sH row stride (bank-conflict-free for WMMA B fetch)
#define NKK   9               // K chunks: ceil(36/4)

// Fused down-sample: replicate-pad, 5x5 Gaussian (x36), stride 2.
// Horizontal pass scalar into LDS; vertical pass via V_WMMA_F32_16X16X4_F32
// with the stride-2 5-tap filter encoded in a 16x36 A matrix (LDS table).
__global__ __launch_bounds__(64) void pyr_down(const float* __restrict__ in,
                                               float* __restrict__ out,
                                               int inH, int inW,
                                               int outH, int outW)
{
    __shared__ float sIn[TINH][INSTR];
    __shared__ float sH[TINH + 1][HSTR];   // +1 zero row for K-padding (j=35)
    __shared__ v2f   sA[NKK][32];          // per-lane A-matrix fragments

    const int tid = threadIdx.x;
    const int b   = blockIdx.z;
    const int oy0 = blockIdx.y * OTH;
    const int ox0 = blockIdx.x * OTW;
    const int iy0 = 2 * oy0;
    const int ix0 = 2 * ox0;

    // separable weights: 6*exp(-x^2/(2*1.08^2)); 6*6 == KSCALE(36)
    // (hipcc constant-folds these expf calls)
    float w[KS];
#pragma unroll
    for (int t = 0; t < KS; ++t) {
        float x = (float)(t - 2);
        w[t] = 6.0f * expf(-x * x * (1.0f / (2.0f * 1.08f * 1.08f)));
    }

    // ---- build A fragments once per block: A[r][j] = w[j-2r], 0<=j-2r<5 ----
    for (int idx = tid; idx < NKK * 32; idx += 64) {
        int kk = idx >> 5;
        int l  = idx & 31;
        int mm = l & 15;
        int ka = 4 * kk + (l >> 4) * 2;   // K index of VGPR0 slot (VGPR1 = ka+1)
        int d0 = ka - 2 * mm;
        int d1 = d0 + 1;
        v2f a;
        a.x = ((unsigned)d0 < 5u) ? w[min(max(d0, 0), 4)] : 0.0f;
        a.y = ((unsigned)d1 < 5u) ? w[min(max(d1, 0), 4)] : 0.0f;
        sA[kk][l] = a;
    }
    if (tid < OTW) sH[TINH][tid] = 0.0f;   // keep zero-weight K column NaN-free

    // ---- stage input tile (replicate clamp), no per-iter division ----
    const float* inb = in + (size_t)b * inH * inW;
    {
        int j = tid / TINW;
        int c = tid - j * TINW;
        for (int idx = tid; idx < TINH * TINW; idx += 64) {
            int gy = iy0 - PAD + j; gy = min(max(gy, 0), inH - 1);
            int gx = ix0 - PAD + c; gx = min(max(gx, 0), inW - 1);
            sIn[j][c] = inb[(size_t)gy * inW + gx];
            c += 64;
            if (c >= TINW) { c -= TINW; ++j; }
        }
    }
    __syncthreads();

    // ---- horizontal 5-tap, stride 2 ----
    for (int idx = tid; idx < TINH * OTW; idx += 64) {
        int j = idx >> 5;          // OTW == 32
        int c = idx & 31;
        float acc = 0.0f;
#pragma unroll
        for (int t = 0; t < KS; ++t) acc += w[t] * sIn[j][2 * c + t];
        sH[j][c] = acc;
    }
    __syncthreads();

    // ---- vertical pass as WMMA: D(16x16) = A(16x36)*H(36x16) ----
    const int lane = tid & 31;
    const int wid  = tid >> 5;       // wave id within block: column half
    const int c0   = wid * 16;
    const int m    = lane & 15;      // N index (B/D)
    const int hi   = lane >> 4;      // 0: lanes 0-15, 1: lanes 16-31

    v8f acc = {};
#pragma unroll
    for (int kk = 0; kk < NKK; ++kk) {
        const int ka = 4 * kk + hi * 2;
        v2f a = sA[kk][lane];
        v2f bm;
        bm.x = sH[ka][c0 + m];
        bm.y = sH[ka + 1][c0 + m];
        acc = __builtin_amdgcn_wmma_f32_16x16x4_f32(false, a, false, bm,
                                                    (short)0, acc, false, false);
    }

    // D layout: VGPR v -> row v (lanes 0-15) / row v+8 (lanes 16-31), col = lane%16
    float* outb = out + (size_t)b * outH * outW;
#pragma unroll
    for (int v = 0; v < 8; ++v) {
        int row = oy0 + v + hi * 8;
        int col = ox0 + c0 + m;
        outb[(size_t)row * outW + col] = acc[v];
    }
}

// 4 intrinsics matrices: K, K*mask, K*mask^2, K*mask^3 (elementwise)
__global__ void intr_kernel(const float* __restrict__ K, float* __restrict__ out)
{
    int t = threadIdx.x;
    if (t >= 9) return;
    const float mask[9] = {0.5f, 1.0f, 0.5f,
                           1.0f, 0.5f, 0.5f,
                           1.0f, 1.0f, 1.0f};
    float v = K[t];
    float msk = mask[t];
    out[t] = v;
    v *= msk; out[ 9 + t] = v;
    v *= msk; out[18 + t] = v;
    v *= msk; out[27 + t] = v;
}

extern "C" void kernel_launch(void* const* d_in, const int* in_sizes, int n_in,
                              void* d_out, int out_size, void* d_ws, size_t ws_size,
                              hipStream_t stream)
{
    (void)in_sizes; (void)n_in; (void)out_size; (void)d_ws; (void)ws_size;

    const float* img = (const float*)d_in[0];
    const float* K   = (const float*)d_in[1];
    float* out = (float*)d_out;

    const int B = 16;
    const size_t L0 = (size_t)B * 1024 * 1280;   // 20,971,520
    const size_t L1 = (size_t)B * 512 * 640;     //  5,242,880
    const size_t L2 = (size_t)B * 256 * 320;     //  1,310,720
    const size_t L3 = (size_t)B * 128 * 160;     //    327,680

    float* o0 = out;
    float* o1 = o0 + L0;
    float* o2 = o1 + L1;
    float* o3 = o2 + L2;
    float* oi = o3 + L3;

    // level 0: straight copy of the input
    hipMemcpyAsync(o0, img, L0 * sizeof(float), hipMemcpyDeviceToDevice, stream);

    dim3 blk(64);
    pyr_down<<<dim3(640/OTW, 512/OTH, B), blk, 0, stream>>>(img, o1, 1024, 1280, 512, 640);
    pyr_down<<<dim3(320/OTW, 256/OTH, B), blk, 0, stream>>>(o1,  o2,  512,  640, 256, 320);
    pyr_down<<<dim3(160/OTW, 128/OTH, B), blk, 0, stream>>>(o2,  o3,  256,  320, 128, 160);

    intr_kernel<<<1, 16, 0, stream>>>(K, oi);
}